// ODAPIDetectionGenerator_47519518163336
// MI455X (gfx1250) — compile-verified
//
#include <hip/hip_runtime.h>
#include <hip/hip_bf16.h>
#include <math.h>

#define BB 32
#define HH 128
#define WW 128
#define CC 90
#define KTOP 100
#define NBINS 16384
#define CAND_MAX 4096
#define CH 15                      // channel chunk per stage
#define NCHUNK (CC / CH)           // 6
#define TROW (WW * CH)             // 1920 floats per row-slab
#define TELEMS (3 * TROW)          // 5760 floats = 23040 B per buffer
#define PEAK_EPS 1e-6f

#define HIST_BYTES ((size_t)BB * NBINS * 4)   // 2 MB
// ws layout: [hist 2MB][cnt 32*u32][pad][thr 32*u32][pad][cand 32*4096*u64]
#define WS_CNT_OFF  HIST_BYTES
#define WS_THR_OFF  (HIST_BYTES + 128)
#define WS_CAND_OFF (HIST_BYTES + 256)

// ---------------- CDNA5 feature detection ----------------
#if defined(__HIP_DEVICE_COMPILE__) && defined(__gfx1250__) && \
    __has_builtin(__builtin_amdgcn_tensor_load_to_lds)
#define USE_TDM 1
#else
#define USE_TDM 0
#endif

#if defined(__HIP_DEVICE_COMPILE__) && defined(__gfx1250__) && \
    __has_builtin(__builtin_amdgcn_global_load_async_to_lds_b32)
#define USE_ASYNC_LDS 1
#else
#define USE_ASYNC_LDS 0
#endif

typedef unsigned int u32x4 __attribute__((ext_vector_type(4)));
typedef int i32x8 __attribute__((ext_vector_type(8)));
typedef int i32x4 __attribute__((ext_vector_type(4)));
typedef __attribute__((address_space(1))) int as1_int;
typedef __attribute__((address_space(3))) int as3_int;

__device__ __forceinline__ void async_ld_f32(const float* g, float* l) {
#if USE_ASYNC_LDS
  __builtin_amdgcn_global_load_async_to_lds_b32(
      (as1_int*)(unsigned long long)(const void*)g,
      (as3_int*)(unsigned int)(unsigned long long)(const void*)l,
      0, 0);
#else
  *l = *g;
#endif
}

// Stage one chunk: nrows rows of [WW x CH] (x-stride CC, row-stride WW*CC
// elements) into LDS at buf + rbase*TROW. TDM: one instruction from wave 0.
__device__ __forceinline__ void stage_chunk(const float* __restrict__ hm,
                                            float* buf, int b, int row0,
                                            int nrows, int rbase, int c0,
                                            int tid) {
#if USE_TDM
  if (tid < 32) {
    const float* gbase = hm + (((size_t)b * HH + row0) * WW) * CC + c0;
    unsigned long long ga = (unsigned long long)(const void*)gbase;
    unsigned lds_off = (unsigned)(unsigned long long)(const void*)(buf + rbase * TROW);
    // D# group 0: count=1 | lds_addr | global_addr(57b) | type=2
    u32x4 g0 = {0x1u, lds_off, (unsigned)(ga & 0xFFFFFFFFu),
                (unsigned)((ga >> 32) & 0x01FFFFFFu) | 0x80000000u};
    // D# group 1: data_size=4B; tensor_dim0=CH, dim1=WW; tile CHxWWxnrows;
    // dim0_stride=CC, dim1_stride=WW*CC (element units)
    i32x8 g1 = {
        0x20000,                                        // data_size=2 (4B)
        (int)((unsigned)CH << 16),                      // tensor_dim0[15:0]
        (int)((unsigned)WW << 16),                      // dim0 hi=0 | dim1 lo
        (int)((unsigned)CH << 16),                      // dim1 hi=0 | tile_dim0
        (int)((unsigned)WW | ((unsigned)nrows << 16)),  // tile_dim1 | tile_dim2
        (int)(CC),                                      // dim0_stride lo
        (int)((unsigned)(WW * CC) << 16),               // d0s hi=0 | d1s lo
        (int)((unsigned)(WW * CC) >> 16)                // d1s hi
    };
    i32x4 g2 = {nrows, 0, 0, 0};                        // tensor_dim2
    i32x4 g3 = {0, 0, 0, 0};
    i32x8 g4 = {0, 0, 0, 0, 0, 0, 0, 0};                // 6-arg form: extra group
    __builtin_amdgcn_tensor_load_to_lds(g0, g1, g2, g3, g4, 0);
  }
#else
  for (int i = tid; i < nrows * TROW; i += 256) {
    int r = i / TROW;
    int rem = i - r * TROW;
    int x = rem / CH;
    int cc = rem - x * CH;
    const float* src = hm + (((size_t)b * HH + (row0 + r)) * WW + x) * CC + (c0 + cc);
    async_ld_f32(src, buf + (rbase + r) * TROW + rem);
  }
#endif
}

__device__ __forceinline__ void stage_wait(int tid) {
#if USE_TDM
  if (tid < 32) {
#if __has_builtin(__builtin_amdgcn_s_wait_tensorcnt)
    __builtin_amdgcn_s_wait_tensorcnt(0);
#else
    asm volatile("s_wait_tensorcnt 0" ::: "memory");
#endif
  }
#elif USE_ASYNC_LDS
#if __has_builtin(__builtin_amdgcn_s_wait_asynccnt)
  __builtin_amdgcn_s_wait_asynccnt(0);
#else
  asm volatile("s_wait_asynccnt 0" ::: "memory");
#endif
#endif
  (void)tid;
}

__device__ __forceinline__ float sigmoidf_(float x) {
  return 1.0f / (1.0f + __expf(-x));
}

// ---------------- Pass kernel: peaks -> histogram (mode 0) or compact (mode 1)
__global__ void __launch_bounds__(256)
peaks_pass(const float* __restrict__ hm,
           unsigned* __restrict__ hist,
           const unsigned* __restrict__ thr,
           unsigned* __restrict__ cnt,
           unsigned long long* __restrict__ cand,
           int mode) {
  __shared__ float tile[2][TELEMS];  // double buffer, 46080 B total
  const int y = blockIdx.x;
  const int b = blockIdx.y;
  const int tid = (int)threadIdx.x;
  const unsigned tb = (mode != 0) ? thr[b] : 0u;

  const int row0 = (y == 0) ? 0 : y - 1;
  const int row1 = (y == HH - 1) ? HH - 1 : y + 1;
  const int nrows = row1 - row0 + 1;
  const int rbase = row0 - (y - 1);  // slot of first valid row; center = slot 1
  const bool edge = (y == 0) || (y == HH - 1);

  stage_chunk(hm, tile[0], b, row0, nrows, rbase, 0, tid);

  for (int ci = 0; ci < NCHUNK; ++ci) {
    const int buf = ci & 1;
    const int c0 = ci * CH;
    stage_wait(tid);
    __syncthreads();  // chunk ci resident in tile[buf]

    // Overlap: wave0 kicks off the DMA for chunk ci+1 into the other buffer.
    if (ci + 1 < NCHUNK)
      stage_chunk(hm, tile[buf ^ 1], b, row0, nrows, rbase, c0 + CH, tid);

    // SAME padding rows (edge blocks only): -inf never wins the max.
    if (edge) {
      if (y == 0)
        for (int i = tid; i < TROW; i += 256) tile[buf][i] = -INFINITY;
      else
        for (int i = tid; i < TROW; i += 256) tile[buf][2 * TROW + i] = -INFINITY;
      __syncthreads();
    }

    // 3x3 max pool on logits (sigmoid is monotone), peak test, emit.
    for (int i = tid; i < TROW; i += 256) {
      int x  = i / CH;
      int cc = i - x * CH;
      float center = tile[buf][TROW + i];  // slot 1 is row y
      float m = center;
#pragma unroll
      for (int r = 0; r < 3; ++r) {
        int rb = r * TROW + cc;
        if (x > 0)      m = fmaxf(m, tile[buf][rb + (x - 1) * CH]);
        m = fmaxf(m, tile[buf][rb + x * CH]);
        if (x < WW - 1) m = fmaxf(m, tile[buf][rb + (x + 1) * CH]);
      }
      float prob   = sigmoidf_(center);
      float pooled = sigmoidf_(m);
      if (fabsf(prob - pooled) < PEAK_EPS) {
        unsigned bits = __float_as_uint(prob);
        if (mode == 0) {
          atomicAdd(&hist[(size_t)b * NBINS + (bits >> 18)], 1u);
        } else if ((bits >> 18) >= tb) {
          unsigned pos = atomicAdd(&cnt[b], 1u);
          if (pos < CAND_MAX) {
            unsigned flat = ((unsigned)y * WW + (unsigned)x) * CC + (unsigned)(c0 + cc);
            // sort key: score desc, then lower flat index first (jax tie rule)
            cand[(size_t)b * CAND_MAX + pos] =
                ((unsigned long long)bits << 32) | (0xFFFFFFFFu - flat);
          }
        }
      }
    }
    __syncthreads();  // done with tile[buf] before chunk ci+2 overwrites it
  }
}

// ---------------- Per-batch threshold bin from histogram ----------------
__global__ void __launch_bounds__(256)
find_thresh(const unsigned* __restrict__ hist, unsigned* __restrict__ thr) {
  __shared__ unsigned part[256];
  const int b = blockIdx.x;
  const int tid = (int)threadIdx.x;
  const unsigned* h = hist + (size_t)b * NBINS;
  const int chunk = NBINS / 256;  // 64 bins per thread
  __builtin_prefetch(h + tid * chunk, 0, 0);  // global_prefetch_b8
  unsigned s = 0;
  for (int i = 0; i < chunk; ++i) s += h[tid * chunk + i];
  part[tid] = s;
  __syncthreads();
  if (tid == 0) {
    unsigned cum = 0, t = 0;
    for (int p = 255; p >= 0; --p) {
      if (cum + part[p] >= (unsigned)KTOP) {
        for (int bin = (p + 1) * chunk - 1; bin >= p * chunk; --bin) {
          cum += h[bin];
          if (cum >= (unsigned)KTOP) { t = (unsigned)bin; break; }
        }
        break;
      }
      cum += part[p];
    }
    thr[b] = t;  // 0 if fewer than KTOP peaks -> keep everything
  }
}

// ---------------- Sort candidates, decode boxes, write outputs ----------------
__global__ void __launch_bounds__(256)
finalize(const float* __restrict__ ct_size,
         const float* __restrict__ ct_offset,
         const unsigned* __restrict__ cnt,
         const unsigned long long* __restrict__ cand,
         float* __restrict__ out) {
  __shared__ unsigned long long keys[CAND_MAX];  // 32 KB
  __shared__ unsigned ndet;
  const int b = blockIdx.x;
  const int tid = (int)threadIdx.x;
  unsigned n = cnt[b];
  if (n > CAND_MAX) n = CAND_MAX;
  for (int i = tid; i < CAND_MAX; i += 256)
    keys[i] = (i < (int)n) ? cand[(size_t)b * CAND_MAX + i] : 0ull;
  if (tid == 0) ndet = 0;
  __syncthreads();

  // Bitonic sort, descending (zeros sink to the end).
  for (int k = 2; k <= CAND_MAX; k <<= 1) {
    for (int j = k >> 1; j > 0; j >>= 1) {
      for (int i = tid; i < CAND_MAX; i += 256) {
        int ixj = i ^ j;
        if (ixj > i) {
          unsigned long long a = keys[i], c = keys[ixj];
          bool doswap = ((i & k) == 0) ? (a < c) : (a > c);
          if (doswap) { keys[i] = c; keys[ixj] = a; }
        }
      }
      __syncthreads();
    }
  }

  if (tid < KTOP) {
    unsigned long long kk = keys[tid];
    float score = __uint_as_float((unsigned)(kk >> 32));
    unsigned flat = 0xFFFFFFFFu - (unsigned)(kk & 0xFFFFFFFFu);
    if (kk == 0ull) { score = 0.0f; flat = 0u; }
    unsigned q  = flat / CC;        // y*W + x
    unsigned c  = flat - q * CC;    // channel
    unsigned yi = q / WW;
    unsigned xi = q - yi * WW;
    const float* hwp = ct_size   + ((size_t)b * HH * WW + q) * 2;
    const float* ofp = ct_offset + ((size_t)b * HH * WW + q) * 2;
    float hgt = fmaxf(hwp[0], 0.0f);
    float wid = fmaxf(hwp[1], 0.0f);
    float yo = ofp[0], xo = ofp[1];
    float yf = (float)yi, xf = (float)xi;
    float ymin = fminf(fmaxf(yf + yo - 0.5f * hgt, 0.0f), (float)HH);
    float xmin = fminf(fmaxf(xf + xo - 0.5f * wid, 0.0f), (float)WW);
    float ymax = fminf(fmaxf(yf + yo + 0.5f * hgt, 0.0f), (float)HH);
    float xmax = fminf(fmaxf(xf + xo + 0.5f * wid, 0.0f), (float)WW);
    const float sc = 4.0f / 512.0f;  // STRIDE / INPUT_SIZE
    float* bx = out + ((size_t)b * KTOP + tid) * 4;
    bx[0] = fminf(fmaxf(ymin * sc, 0.0f), 1.0f);
    bx[1] = fminf(fmaxf(xmin * sc, 0.0f), 1.0f);
    bx[2] = fminf(fmaxf(ymax * sc, 0.0f), 1.0f);
    bx[3] = fminf(fmaxf(xmax * sc, 0.0f), 1.0f);
    out[(size_t)BB * KTOP * 4 + (size_t)b * KTOP + tid] = (float)c;   // ch_idx
    out[(size_t)BB * KTOP * 5 + (size_t)b * KTOP + tid] = score;      // scores
    if (score > 0.0f) atomicAdd(&ndet, 1u);
  }
  __syncthreads();
  if (tid == 0) out[(size_t)BB * KTOP * 6 + b] = (float)ndet;         // num_dets
}

extern "C" void kernel_launch(void* const* d_in, const int* in_sizes, int n_in,
                              void* d_out, int out_size, void* d_ws, size_t ws_size,
                              hipStream_t stream) {
  (void)in_sizes; (void)n_in; (void)out_size; (void)ws_size;
  const float* hm = (const float*)d_in[0];
  const float* sz = (const float*)d_in[1];
  const float* of = (const float*)d_in[2];
  float* out = (float*)d_out;
  char* ws = (char*)d_ws;
  unsigned* hist = (unsigned*)ws;
  unsigned* cnt  = (unsigned*)(ws + WS_CNT_OFF);
  unsigned* thr  = (unsigned*)(ws + WS_THR_OFF);
  unsigned long long* cand = (unsigned long long*)(ws + WS_CAND_OFF);

  (void)hipMemsetAsync(ws, 0, HIST_BYTES + 128, stream);  // hist + cand counters

  dim3 grid(HH, BB);  // one block per (row, batch)
  peaks_pass<<<grid, 256, 0, stream>>>(hm, hist, thr, cnt, cand, 0);
  find_thresh<<<BB, 256, 0, stream>>>(hist, thr);
  peaks_pass<<<grid, 256, 0, stream>>>(hm, hist, thr, cnt, cand, 1);
  finalize<<<BB, 256, 0, stream>>>(sz, of, cnt, cand, out);
}